// SelectiveMultiHopGCN_59115929862451
// MI455X (gfx1250) — compile-verified
//
#include <hip/hip_runtime.h>
#include <math.h>

#define HIDDEN 128
#define IN_DIM 256

typedef float v2f __attribute__((ext_vector_type(2)));
typedef float v8f __attribute__((ext_vector_type(8)));

// ------------------------------------------------------------------
// Pack W[K x 128] into WMMA B-fragment order:
//   Bp[(k4*8 + n)*32 + lane] = { W[k4*4 + 2*hi][n*16+l15], W[k4*4 + 2*hi + 1][n*16+l15] }
// so the GEMM inner loop does one coalesced global_load_b64 per fragment.
// ------------------------------------------------------------------
__global__ void pack_b_kernel(const float* __restrict__ B, float* __restrict__ Bp, int K)
{
  const int i = blockIdx.x * blockDim.x + threadIdx.x;
  const int total = (K >> 2) * 8 * 32;
  if (i >= total) return;
  const int lane  = i & 31;
  const int n     = (i >> 5) & 7;
  const int kstep = i >> 8;
  const int hi = lane >> 4, l15 = lane & 15;
  const int krow = kstep * 4 + 2 * hi;
  const int col  = n * 16 + l15;
  Bp[2 * i]     = B[(size_t)krow * HIDDEN + col];
  Bp[2 * i + 1] = B[(size_t)(krow + 1) * HIDDEN + col];
}

// ------------------------------------------------------------------
// GEMM: out[M x 128] = A[M x K] @ B[K x 128]  via V_WMMA_F32_16X16X4_F32
// One wave computes a 16-row x 128-col strip: 1 A-frag reused across 8
// accumulators; all 8 B fragments (packed, coalesced b64) preloaded per
// k-step so the 8 WMMAs issue back-to-back after a single wait.
// MODE 0: out0 = relu(acc + bias)                        (input layer)
// MODE 1: out0 = acc (raw hl);  out1 = dinv[row]^2*acc + bias (GCN self-loop init)
// ------------------------------------------------------------------
template <int MODE>
__global__ __launch_bounds__(128, 1)
void gemm_n128(const float* __restrict__ A, const v2f* __restrict__ Bp,
               const float* __restrict__ bias, float* __restrict__ out0,
               float* __restrict__ out1, const float* __restrict__ dinv,
               int M, int K)
{
  const int wave = (int)((blockIdx.x * blockDim.x + threadIdx.x) >> 5);
  const int lane = threadIdx.x & 31;
  const int m0   = wave << 4;
  if (m0 >= M) return;                       // wave-uniform: EXEC stays all-1s
  const int hi  = lane >> 4;                 // 0: K,K+1   1: K+2,K+3
  const int l15 = lane & 15;

  const float* Al = A + (size_t)(m0 + l15) * K + (hi << 1);
  const v2f*   Bl = Bp + lane;

  v8f acc[8] = {};
  const int ksteps = K >> 2;
  for (int k4 = 0; k4 < ksteps; ++k4) {
    const v2f a = *reinterpret_cast<const v2f*>(Al + k4 * 4);
    __builtin_prefetch(Bl + (size_t)(k4 + 8) * 8 * 32, 0, 3);
    v2f b[8];
#pragma unroll
    for (int n = 0; n < 8; ++n) b[n] = Bl[(size_t)(k4 * 8 + n) * 32];
#pragma unroll
    for (int n = 0; n < 8; ++n) {
      acc[n] = __builtin_amdgcn_wmma_f32_16x16x4_f32(false, a, false, b[n],
                                                     (short)0, acc[n], false, false);
    }
  }

  // C/D layout: VGPR r -> row m0 + r + 8*hi, col = n*16 + l15
#pragma unroll
  for (int r = 0; r < 8; ++r) {
    const int row = m0 + r + (hi << 3);
    float dv2 = 0.0f;
    if (MODE == 1) { const float d = dinv[row]; dv2 = d * d; }
#pragma unroll
    for (int n = 0; n < 8; ++n) {
      const int col = n * 16 + l15;
      const float v = acc[n][r];
      if (MODE == 0) {
        out0[(size_t)row * HIDDEN + col] = fmaxf(v + bias[col], 0.0f);
      } else {
        out0[(size_t)row * HIDDEN + col] = v;
        out1[(size_t)row * HIDDEN + col] = fmaf(dv2, v, bias[col]);
      }
    }
  }
}

// ---------------- importance MLP: imp = sigmoid(relu(h@W1+b1)@W2+b2) --------
__global__ void imp_kernel(const float* __restrict__ h, const float* __restrict__ W1,
                           const float* __restrict__ b1, const float* __restrict__ W2,
                           const float* __restrict__ b2, float* __restrict__ imp, int N)
{
  __shared__ float sh[HIDDEN];
  __shared__ float red[64];
  const int n = blockIdx.x;
  const int t = threadIdx.x;                 // 64 threads
  if (n >= N) return;
  sh[t]      = h[(size_t)n * HIDDEN + t];
  sh[t + 64] = h[(size_t)n * HIDDEN + 64 + t];
  __syncthreads();
  float acc = b1[t];
#pragma unroll 8
  for (int i = 0; i < HIDDEN; ++i) acc = fmaf(sh[i], W1[(size_t)i * 64 + t], acc);
  red[t] = fmaxf(acc, 0.0f) * W2[t];
  __syncthreads();
  for (int s = 32; s > 0; s >>= 1) {
    if (t < s) red[t] += red[t + s];
    __syncthreads();
  }
  if (t == 0) imp[n] = 1.0f / (1.0f + expf(-(red[0] + b2[0])));
}

// ---------------- zero deg + stats ------------------------------------------
__global__ void init_kernel(float* __restrict__ deg, double* __restrict__ stats, int N)
{
  const int i = blockIdx.x * blockDim.x + threadIdx.x;
  if (i < N) deg[i] = 0.0f;
  if (i < 2) stats[i] = 0.0;
}

// ---------------- edge weight sum / sumsq (double atomics) ------------------
__global__ void edge_stats_kernel(const int* __restrict__ ei, const float* __restrict__ imp,
                                  double* __restrict__ stats, int E)
{
  __shared__ double ss[256], sq[256];
  const int t = threadIdx.x;
  double s = 0.0, s2 = 0.0;
  for (int e = blockIdx.x * blockDim.x + t; e < E; e += gridDim.x * blockDim.x) {
    const float ew = 0.5f * (imp[ei[e]] + imp[ei[E + e]]);
    s += (double)ew;
    s2 += (double)ew * (double)ew;
  }
  ss[t] = s; sq[t] = s2;
  __syncthreads();
  for (int k = 128; k > 0; k >>= 1) {
    if (t < k) { ss[t] += ss[t + k]; sq[t] += sq[t + k]; }
    __syncthreads();
  }
  if (t == 0) { atomicAdd(&stats[0], ss[0]); atomicAdd(&stats[1], sq[0]); }
}

// ---------------- threshold = mean + unbiased std ---------------------------
__global__ void thr_kernel(const double* __restrict__ stats, float* __restrict__ thr, int E)
{
  const double sum = stats[0], sumsq = stats[1];
  const double mean = sum / (double)E;
  double var = (sumsq - sum * sum / (double)E) / (double)(E - 1);
  if (var < 0.0) var = 0.0;
  thr[0] = (float)(mean + sqrt(var));
}

// ---------------- masked degree: deg[col] += (ew > thr) ---------------------
__global__ void degree_kernel(const int* __restrict__ ei, const float* __restrict__ imp,
                              const float* __restrict__ thr, float* __restrict__ deg, int E)
{
  const int e = blockIdx.x * blockDim.x + threadIdx.x;
  if (e >= E) return;
  const int r = ei[e], c = ei[E + e];
  const float ew = 0.5f * (imp[r] + imp[c]);
  if (ew > thr[0]) atomicAdd(&deg[c], 1.0f);
}

// ---------------- dinv = rsqrt(deg + 1) -------------------------------------
__global__ void dinv_kernel(const float* __restrict__ deg, float* __restrict__ dinv, int N)
{
  const int i = blockIdx.x * blockDim.x + threadIdx.x;
  if (i < N) dinv[i] = rsqrtf(deg[i] + 1.0f);
}

// ---------------- masked scatter: agg[col] += dinv[r]*dinv[c]*hl[row] -------
// one wave per edge; lane handles 4 contiguous floats (float4 gather + 4 atomics)
__global__ void scatter_kernel(const int* __restrict__ ei, const float* __restrict__ imp,
                               const float* __restrict__ thr, const float* __restrict__ dinv,
                               const float* __restrict__ hl, float* __restrict__ agg, int E)
{
  const int e    = (int)((blockIdx.x * blockDim.x + threadIdx.x) >> 5);
  const int lane = threadIdx.x & 31;
  if (e >= E) return;
  const int r = ei[e], c = ei[E + e];
  const float ew = 0.5f * (imp[r] + imp[c]);
  if (!(ew > thr[0])) return;                 // wave-uniform skip (~85% of edges)
  const float w = dinv[r] * dinv[c];
  const float4 v = reinterpret_cast<const float4*>(hl + (size_t)r * HIDDEN)[lane];
  float* dst = agg + (size_t)c * HIDDEN + lane * 4;
  atomicAdd(dst + 0, w * v.x);
  atomicAdd(dst + 1, w * v.y);
  atomicAdd(dst + 2, w * v.z);
  atomicAdd(dst + 3, w * v.w);
}

// ---------------- elementwise relu (in place) -------------------------------
__global__ void relu_kernel(float* __restrict__ a, size_t n)
{
  const size_t i = (size_t)blockIdx.x * blockDim.x + threadIdx.x;
  if (i < n) a[i] = fmaxf(a[i], 0.0f);
}

// ---------------- out[n] = concat(h,c1,c2) . W_out + b ----------------------
__global__ void out_kernel(const float* __restrict__ h, const float* __restrict__ c1,
                           const float* __restrict__ c2, const float* __restrict__ Wout,
                           const float* __restrict__ bout, float* __restrict__ out, int N)
{
  const int n    = (int)((blockIdx.x * blockDim.x + threadIdx.x) >> 5);
  const int lane = threadIdx.x & 31;
  if (n >= N) return;
  const float* segs[3] = { h, c1, c2 };
  float acc = 0.0f;
#pragma unroll
  for (int s = 0; s < 3; ++s) {
    const float* base = segs[s] + (size_t)n * HIDDEN;
    const float* w    = Wout + s * HIDDEN;
#pragma unroll
    for (int j = 0; j < HIDDEN; j += 32) acc = fmaf(base[j + lane], w[j + lane], acc);
  }
  for (int m = 16; m > 0; m >>= 1) acc += __shfl_xor(acc, m, 32);
  if (lane == 0) out[n] = acc + bout[0];
}

// ---------------------------------------------------------------------------
extern "C" void kernel_launch(void* const* d_in, const int* in_sizes, int n_in,
                              void* d_out, int out_size, void* d_ws, size_t ws_size,
                              hipStream_t stream)
{
  const float* x      = (const float*)d_in[0];
  const int*   eidx   = (const int*)  d_in[1];    // [2, E] row-major: rows then cols
  const float* W_in   = (const float*)d_in[2];
  const float* b_in   = (const float*)d_in[3];
  const float* W_imp1 = (const float*)d_in[4];
  const float* b_imp1 = (const float*)d_in[5];
  const float* W_imp2 = (const float*)d_in[6];
  const float* b_imp2 = (const float*)d_in[7];
  const float* W_conv = (const float*)d_in[8];    // [2,128,128]
  const float* b_conv = (const float*)d_in[9];    // [2,128]
  const float* W_out  = (const float*)d_in[10];   // [384,1]
  const float* b_out  = (const float*)d_in[11];
  float*       out    = (float*)d_out;

  const int N = in_sizes[0] / IN_DIM;
  const int E = in_sizes[1] / 2;

  // workspace carve-out (256B aligned)
  size_t off = 0;
  auto carve = [&](size_t bytes) { size_t o = off; off += (bytes + 255) & ~(size_t)255; return o; };
  char* ws = (char*)d_ws;
  const size_t featB = (size_t)N * HIDDEN * sizeof(float);
  float*  h     = (float*) (ws + carve(featB));
  float*  hl    = (float*) (ws + carve(featB));
  float*  c1    = (float*) (ws + carve(featB));   // agg buffer hop 0 -> feat 1
  float*  c2    = (float*) (ws + carve(featB));   // agg buffer hop 1 -> feat 2
  float*  imp   = (float*) (ws + carve((size_t)N * sizeof(float)));
  float*  dinv  = (float*) (ws + carve((size_t)N * sizeof(float)));
  float*  deg   = (float*) (ws + carve((size_t)N * sizeof(float)));
  double* stats = (double*)(ws + carve(2 * sizeof(double)));
  float*  thr   = (float*) (ws + carve(sizeof(float)));
  float*  BpIn  = (float*) (ws + carve((size_t)IN_DIM * HIDDEN * sizeof(float)));
  float*  BpC0  = (float*) (ws + carve((size_t)HIDDEN * HIDDEN * sizeof(float)));
  float*  BpC1  = (float*) (ws + carve((size_t)HIDDEN * HIDDEN * sizeof(float)));
  (void)ws_size; (void)n_in; (void)out_size;

  const int Mtiles   = (N + 15) / 16;             // 16 rows per wave
  const int gemmBlks = (Mtiles + 3) / 4;          // 4 waves / 128-thread block

  // 0) zero deg + stats; pack weights into WMMA B-fragment order
  init_kernel<<<(N + 255) / 256, 256, 0, stream>>>(deg, stats, N);
  pack_b_kernel<<<(IN_DIM / 4 * 8 * 32 + 255) / 256, 256, 0, stream>>>(W_in, BpIn, IN_DIM);
  pack_b_kernel<<<(HIDDEN / 4 * 8 * 32 + 255) / 256, 256, 0, stream>>>(W_conv, BpC0, HIDDEN);
  pack_b_kernel<<<(HIDDEN / 4 * 8 * 32 + 255) / 256, 256, 0, stream>>>(W_conv + HIDDEN * HIDDEN,
                                                                       BpC1, HIDDEN);
  // 1) h = relu(x @ W_in + b_in)       [fp32 WMMA]
  gemm_n128<0><<<gemmBlks, 128, 0, stream>>>(x, (const v2f*)BpIn, b_in, h, nullptr, nullptr,
                                             N, IN_DIM);
  // 2) node importance
  imp_kernel<<<N, 64, 0, stream>>>(h, W_imp1, b_imp1, W_imp2, b_imp2, imp, N);
  // 3) edge-weight mean / std
  edge_stats_kernel<<<1024, 256, 0, stream>>>(eidx, imp, stats, E);
  thr_kernel<<<1, 1, 0, stream>>>(stats, thr, E);
  // 4) masked degree + dinv
  degree_kernel<<<(E + 255) / 256, 256, 0, stream>>>(eidx, imp, thr, deg, E);
  dinv_kernel<<<(N + 255) / 256, 256, 0, stream>>>(deg, dinv, N);

  const size_t scatterBlks = ((size_t)E * 32 + 255) / 256;
  const size_t reluBlks    = ((size_t)N * HIDDEN + 255) / 256;

  // hop 0: hl = h @ W_conv[0];  c1 = dinv^2*hl + b (self-loop init); scatter; relu
  gemm_n128<1><<<gemmBlks, 128, 0, stream>>>(h, (const v2f*)BpC0, b_conv, hl, c1, dinv,
                                             N, HIDDEN);
  scatter_kernel<<<(int)scatterBlks, 256, 0, stream>>>(eidx, imp, thr, dinv, hl, c1, E);
  relu_kernel<<<(int)reluBlks, 256, 0, stream>>>(c1, (size_t)N * HIDDEN);

  // hop 1: same on c1
  gemm_n128<1><<<gemmBlks, 128, 0, stream>>>(c1, (const v2f*)BpC1, b_conv + HIDDEN,
                                             hl, c2, dinv, N, HIDDEN);
  scatter_kernel<<<(int)scatterBlks, 256, 0, stream>>>(eidx, imp, thr, dinv, hl, c2, E);
  relu_kernel<<<(int)reluBlks, 256, 0, stream>>>(c2, (size_t)N * HIDDEN);

  // 5) out = concat(h,c1,c2) @ W_out + b_out
  out_kernel<<<(N * 32 + 255) / 256, 256, 0, stream>>>(h, c1, c2, W_out, b_out, out, N);
}